// PerActionQNet_13606456394262
// MI455X (gfx1250) — compile-verified
//
#include <hip/hip_runtime.h>

typedef __attribute__((ext_vector_type(2))) float v2f;
typedef __attribute__((ext_vector_type(4))) float v4f;
typedef __attribute__((ext_vector_type(8))) float v8f;

#define N_MAX 2048
#define M_MAX 2048
#define D_J 16
#define D_M 16
#define D_G 8
#define HIDDEN 64
#define TILE 64
#define PITCH 68  // 64 + 4 floats pad: conflict-reduced, keeps 16B alignment

__device__ __forceinline__ float relu_f(float x) { return fmaxf(x, 0.0f); }

// ---------------- K0: hgb[h] = b1[h] + sum_k relu(g@Wg + bg)[k] * W1g[k][h] ----
__global__ void k0_global_feat(const float* __restrict__ g,
                               const float* __restrict__ Wg,
                               const float* __restrict__ bg,
                               const float* __restrict__ W1,
                               const float* __restrict__ b1,
                               float* __restrict__ hgb) {
    const int h = threadIdx.x;
    if (h >= HIDDEN) return;
    float gf[D_G];
#pragma unroll
    for (int k = 0; k < D_G; ++k) {
        float acc = bg[k];
#pragma unroll
        for (int i = 0; i < D_G; ++i) acc = fmaf(g[i], Wg[i * D_G + k], acc);
        gf[k] = relu_f(acc);
    }
    float acc = b1[h];
#pragma unroll
    for (int k = 0; k < D_G; ++k)
        acc = fmaf(gf[k], W1[(D_J + D_M + k) * HIDDEN + h], acc);
    hgb[h] = acc;
}

// ---------------- K1: feature GEMMs via V_WMMA_F32_16X16X4_F32 ----------------
// One wave (32 lanes) per 16-row tile. hj = relu(job*Wj+bj) @ W1[0:16]
//                                      hmg = relu(mac*Wm+bm) @ W1[16:32] + hgb
__global__ __launch_bounds__(32) void k1_features(
        const float* __restrict__ job, const float* __restrict__ mac,
        const float* __restrict__ Wj,  const float* __restrict__ bj,
        const float* __restrict__ Wm,  const float* __restrict__ bm,
        const float* __restrict__ W1,  const float* __restrict__ hgb,
        float* __restrict__ hj, float* __restrict__ hmg) {
    const int lane  = threadIdx.x;      // 0..31
    const int Mrow  = lane & 15;        // matrix row (A) / col (B,D)
    const int khalf = lane >> 4;        // K-split: lanes 16..31 hold K+2,K+3
    const int tile  = blockIdx.x;       // 0..255
    const bool isJob = tile < (N_MAX / 16);
    const int n0 = (isJob ? tile : tile - (N_MAX / 16)) * 16;

    const float* vec = isJob ? job : mac;
    const float* Wk  = isJob ? Wj : Wm;
    const float* bk  = isJob ? bj : bm;
    const int rowOff = isJob ? 0 : D_J;
    float* outp      = isJob ? hj : hmg;

    // A fragments (16x4 f32, K=16 in 4 steps): a.x=K0+2*khalf, a.y=that+1
    const float x = vec[n0 + Mrow];
    v2f afrag[4];
#pragma unroll
    for (int s = 0; s < 4; ++s) {
        const int k0 = 4 * s + 2 * khalf;
        afrag[s].x = relu_f(fmaf(x, Wk[k0],     bk[k0]));
        afrag[s].y = relu_f(fmaf(x, Wk[k0 + 1], bk[k0 + 1]));
    }

#pragma unroll
    for (int c = 0; c < 4; ++c) {       // 4 chunks of 16 hidden columns
        const int h = c * 16 + Mrow;    // this lane's N / output column
        v8f acc = {};
#pragma unroll
        for (int s = 0; s < 4; ++s) {
            const int k0 = 4 * s + 2 * khalf;
            const float* w1p = W1 + (size_t)(rowOff + k0) * HIDDEN + h;
            v2f bfrag;
            bfrag.x = w1p[0];
            bfrag.y = w1p[HIDDEN];
            acc = __builtin_amdgcn_wmma_f32_16x16x4_f32(
                false, afrag[s], false, bfrag, (short)0, acc, false, false);
        }
        float addv = 0.0f;
        if (!isJob) addv = hgb[h];      // fold hg@W1g + b1 into the mac side
#pragma unroll
        for (int r = 0; r < 8; ++r)     // D vgpr r -> row r + 8*khalf
            outp[(size_t)(n0 + r + 8 * khalf) * HIDDEN + h] = acc[r] + addv;
    }
}

// ---------------- K2: fused Q[n,m] = sum_h relu(hj+hmg)*w2 + b2 ---------------
__global__ __launch_bounds__(256) void k2_main(
        const float* __restrict__ hj, const float* __restrict__ hmg,
        const float* __restrict__ W2, const float* __restrict__ b2,
        float* __restrict__ Q) {
    __shared__ float sA[TILE * PITCH];  // [h][nLocal] (transposed for b128 reads)
    __shared__ float sB[TILE * PITCH];  // [h][mLocal]
    __shared__ float sW2[HIDDEN];

    const int tid = threadIdx.x;
    const int n0 = blockIdx.y * TILE;
    const int m0 = blockIdx.x * TILE;

    for (int idx = tid; idx < TILE * HIDDEN; idx += 256) {
        const int r = idx >> 6;         // local row
        const int h = idx & 63;
        sA[h * PITCH + r] = hj [(size_t)(n0 + r) * HIDDEN + h];
        sB[h * PITCH + r] = hmg[(size_t)(m0 + r) * HIDDEN + h];
    }
    if (tid < HIDDEN) sW2[tid] = W2[tid];
    __syncthreads();

    const int ni = (tid >> 4) * 4;      // 4 job rows per thread
    const int mi = (tid & 15) * 4;      // 4 machine cols per thread

    float acc[4][4] = {};
#pragma unroll 8
    for (int h = 0; h < HIDDEN; ++h) {
        const v4f a = *(const v4f*)(&sA[h * PITCH + ni]);
        const v4f b = *(const v4f*)(&sB[h * PITCH + mi]);
        const float w = sW2[h];
#pragma unroll
        for (int i = 0; i < 4; ++i)
#pragma unroll
            for (int j = 0; j < 4; ++j)
                acc[i][j] = fmaf(relu_f(a[i] + b[j]), w, acc[i][j]);
    }

    const float bias = b2[0];
#pragma unroll
    for (int i = 0; i < 4; ++i) {
        v4f row;
#pragma unroll
        for (int j = 0; j < 4; ++j) row[j] = acc[i][j] + bias;
        __builtin_nontemporal_store(
            row, (v4f*)(&Q[(size_t)(n0 + ni + i) * M_MAX + (m0 + mi)]));
    }
}

// ------------------------------------------------------------------------------
extern "C" void kernel_launch(void* const* d_in, const int* in_sizes, int n_in,
                              void* d_out, int out_size, void* d_ws, size_t ws_size,
                              hipStream_t stream) {
    const float* job = (const float*)d_in[0];
    const float* mac = (const float*)d_in[1];
    const float* g   = (const float*)d_in[2];
    const float* Wj  = (const float*)d_in[3];
    const float* bj  = (const float*)d_in[4];
    const float* Wm  = (const float*)d_in[5];
    const float* bm  = (const float*)d_in[6];
    const float* Wg  = (const float*)d_in[7];
    const float* bg  = (const float*)d_in[8];
    const float* W1  = (const float*)d_in[9];
    const float* b1  = (const float*)d_in[10];
    const float* W2  = (const float*)d_in[11];
    const float* b2  = (const float*)d_in[12];
    float* Q = (float*)d_out;

    float* hj  = (float*)d_ws;                       // 2048*64 f32
    float* hmg = hj  + (size_t)N_MAX * HIDDEN;       // 2048*64 f32
    float* hgb = hmg + (size_t)M_MAX * HIDDEN;       // 64 f32

    k0_global_feat<<<1, 64, 0, stream>>>(g, Wg, bg, W1, b1, hgb);
    k1_features<<<(N_MAX / 16) + (M_MAX / 16), 32, 0, stream>>>(
        job, mac, Wj, bj, Wm, bm, W1, hgb, hj, hmg);
    k2_main<<<dim3(M_MAX / TILE, N_MAX / TILE), 256, 0, stream>>>(
        hj, hmg, W2, b2, Q);
}